// GINLayer_65532611002909
// MI455X (gfx1250) — compile-verified
//
#include <hip/hip_runtime.h>
#include <math.h>

// GINEConv: N=100000 nodes, E=1600000 edges, IN=EMBED=64, EDGE_DIM=16, EPS=0
#define NNODE 100000
#define NEDGE 1600000
#define DIM   64
#define EDIM  16

typedef float v2f __attribute__((ext_vector_type(2)));
typedef float v8f __attribute__((ext_vector_type(8)));

__global__ void gine_zero_kernel(float* __restrict__ p, int n) {
    int i = blockIdx.x * blockDim.x + threadIdx.x;
    if (i < n) p[i] = 0.0f;
}

// ---------------------------------------------------------------------------
// Edge kernel: e = edge_attr @ We + be ; msg = relu(x[src] + e);
//              aggr[dst] += msg   (atomic f32, L2-resident target)
// One wave per 16-edge tile, WMMA f32 16x16x4 chained over K=16.
// ---------------------------------------------------------------------------
__global__ __launch_bounds__(256)
void gine_edge_kernel(const float* __restrict__ x,       // [N,64]
                      const int*   __restrict__ ei,      // [2,E] (src row, dst row)
                      const float* __restrict__ ea,      // [E,16]
                      const float* __restrict__ We,      // [16,64]
                      const float* __restrict__ be,      // [64]
                      float*       __restrict__ aggr)    // [N,64] scratch
{
    const int lane = threadIdx.x & 31;
    const int wave = threadIdx.x >> 5;
    const int half = lane >> 4;      // 0: lanes 0-15, 1: lanes 16-31
    const int l16  = lane & 15;
    const int gwave  = blockIdx.x * (blockDim.x >> 5) + wave;
    const int nwaves = gridDim.x * (blockDim.x >> 5);

    // B fragments of We (4x16 sub-tiles): frag[k][nt]
    // 32-bit B layout: half 0 holds K=4k+0,4k+1 ; half 1 holds K=4k+2,4k+3 ; N = 16*nt + l16
    v2f bw[4][4];
    float bias[4];
#pragma unroll
    for (int k = 0; k < 4; ++k) {
        const int kk = 4 * k + 2 * half;
#pragma unroll
        for (int nt = 0; nt < 4; ++nt) {
            const int c = 16 * nt + l16;
            bw[k][nt].x = We[(kk + 0) * DIM + c];
            bw[k][nt].y = We[(kk + 1) * DIM + c];
        }
    }
#pragma unroll
    for (int nt = 0; nt < 4; ++nt) bias[nt] = be[16 * nt + l16];

    const int* srcArr = ei;
    const int* dstArr = ei + NEDGE;
    const int  ntiles = NEDGE / 16;

    for (int t = gwave; t < ntiles; t += nwaves) {
        const int ebase = t * 16;

        // prefetch next tile's edge_attr rows (global_prefetch_b8)
        const int tn = t + nwaves;
        if (tn < ntiles)
            __builtin_prefetch(ea + (size_t)(tn * 16 + l16) * EDIM, 0, 0);

        // A fragments: row M = ebase + l16 ; half selects K pair within each 4-chunk
        const float* arow = ea + (size_t)(ebase + l16) * EDIM + 2 * half;
        v2f a[4];
#pragma unroll
        for (int k = 0; k < 4; ++k) { a[k].x = arow[4 * k]; a[k].y = arow[4 * k + 1]; }

        v8f acc[4];
#pragma unroll
        for (int nt = 0; nt < 4; ++nt) {
            v8f c = {};
#pragma unroll
            for (int k = 0; k < 4; ++k)
                c = __builtin_amdgcn_wmma_f32_16x16x4_f32(false, a[k], false, bw[k][nt],
                                                          (short)0, c, false, false);
            acc[nt] = c;
        }

        // D layout: element (M = v + 8*half, N = 16*nt + l16) in acc[nt][v]
#pragma unroll
        for (int v = 0; v < 8; ++v) {
            const int m   = v + 8 * half;
            const int s   = srcArr[ebase + m];
            const int d   = dstArr[ebase + m];
            const float* xr = x    + (size_t)s * DIM;
            float*       ar = aggr + (size_t)d * DIM;
#pragma unroll
            for (int nt = 0; nt < 4; ++nt) {
                const int c = 16 * nt + l16;
                float val = acc[nt][v] + bias[nt] + xr[c];
                val = fmaxf(val, 0.0f);
                atomicAdd(&ar[c], val);
            }
        }
    }
}

// ---------------------------------------------------------------------------
// Node kernel: h = x + aggr ; h1 = gelu(h@W1+b1) ; out = h1@W2+b2
// One wave per 16-node tile; per-wave LDS slice redistributes D->A layout.
// ---------------------------------------------------------------------------
__global__ __launch_bounds__(256)
void gine_node_kernel(const float* __restrict__ x,
                      const float* __restrict__ aggr,
                      const float* __restrict__ W1, const float* __restrict__ b1,
                      const float* __restrict__ W2, const float* __restrict__ b2,
                      float* __restrict__ out)
{
    __shared__ float h1s[8][16 * DIM];   // 32 KB: one 16x64 tile per wave

    const int lane = threadIdx.x & 31;
    const int wave = threadIdx.x >> 5;
    const int half = lane >> 4;
    const int l16  = lane & 15;
    const int gwave = blockIdx.x * (blockDim.x >> 5) + wave;
    const int tiles = NNODE / 16;
    if (gwave >= tiles) return;          // wave-uniform
    const int nbase = gwave * 16;

    // A fragments of h = x + aggr (K = 64 -> 16 k-steps)
    v2f a[16];
    {
        const float* xr = x    + (size_t)(nbase + l16) * DIM + 2 * half;
        const float* gr = aggr + (size_t)(nbase + l16) * DIM + 2 * half;
#pragma unroll
        for (int k = 0; k < 16; ++k) {
            a[k].x = xr[4 * k]     + gr[4 * k];
            a[k].y = xr[4 * k + 1] + gr[4 * k + 1];
        }
    }

    float* hs = &h1s[wave][0];

    // ---- layer 1: h1 = gelu(h @ W1 + b1) ----
#pragma unroll
    for (int nt = 0; nt < 4; ++nt) {
        v8f c = {};
        const int col = 16 * nt + l16;
#pragma unroll
        for (int k = 0; k < 16; ++k) {
            const int kk = 4 * k + 2 * half;
            v2f b;
            b.x = W1[(kk + 0) * DIM + col];
            b.y = W1[(kk + 1) * DIM + col];
            c = __builtin_amdgcn_wmma_f32_16x16x4_f32(false, a[k], false, b,
                                                      (short)0, c, false, false);
        }
        const float bb = b1[col];
#pragma unroll
        for (int v = 0; v < 8; ++v) {
            float h = c[v] + bb;
            h = 0.5f * h * (1.0f + erff(h * 0.70710678118654752f));   // exact GELU
            hs[(v + 8 * half) * DIM + col] = h;
        }
    }

    // wave-internal LDS ordering + block compiler reordering across the staging
    asm volatile("s_wait_dscnt 0" ::: "memory");

    // reload A fragments of h1 from the wave's LDS slice
#pragma unroll
    for (int k = 0; k < 16; ++k) {
        const int kk = 4 * k + 2 * half;
        a[k].x = hs[l16 * DIM + kk + 0];
        a[k].y = hs[l16 * DIM + kk + 1];
    }

    // ---- layer 2: out = h1 @ W2 + b2 ----
#pragma unroll
    for (int nt = 0; nt < 4; ++nt) {
        v8f c = {};
        const int col = 16 * nt + l16;
#pragma unroll
        for (int k = 0; k < 16; ++k) {
            const int kk = 4 * k + 2 * half;
            v2f b;
            b.x = W2[(kk + 0) * DIM + col];
            b.y = W2[(kk + 1) * DIM + col];
            c = __builtin_amdgcn_wmma_f32_16x16x4_f32(false, a[k], false, b,
                                                      (short)0, c, false, false);
        }
        const float bb = b2[col];
#pragma unroll
        for (int v = 0; v < 8; ++v)
            out[(size_t)(nbase + v + 8 * half) * DIM + col] = c[v] + bb;
    }
}

extern "C" void kernel_launch(void* const* d_in, const int* in_sizes, int n_in,
                              void* d_out, int out_size, void* d_ws, size_t ws_size,
                              hipStream_t stream) {
    const float* x  = (const float*)d_in[0];
    const int*   ei = (const int*)  d_in[1];   // harness delivers integers as int32
    const float* ea = (const float*)d_in[2];
    const float* We = (const float*)d_in[3];
    const float* be = (const float*)d_in[4];
    const float* W1 = (const float*)d_in[5];
    const float* b1 = (const float*)d_in[6];
    const float* W2 = (const float*)d_in[7];
    const float* b2 = (const float*)d_in[8];
    float* out  = (float*)d_out;
    float* aggr = (float*)d_ws;                // N*64 floats = 25.6 MB scratch

    const int nz = NNODE * DIM;
    gine_zero_kernel<<<(nz + 255) / 256, 256, 0, stream>>>(aggr, nz);

    // 2500 blocks * 8 waves = 20000 wave-tiles covering 100000 edge tiles
    gine_edge_kernel<<<2500, 256, 0, stream>>>(x, ei, ea, We, be, aggr);

    // 6250 node tiles, 8 per block
    gine_node_kernel<<<(NNODE / 16 + 7) / 8, 256, 0, stream>>>(x, aggr, W1, b1, W2, b2, out);
}